// GCNEncoder_61881888800780
// MI455X (gfx1250) — compile-verified
//
#include <hip/hip_runtime.h>
#include <hip/hip_bf16.h>

typedef __attribute__((ext_vector_type(2))) float v2f;
typedef __attribute__((ext_vector_type(8))) float v8f;

#define LN_EPS 1e-5f

// ------------------------------------------------------------------
// GEMM: H[N,128] = A[N,128] @ W[128,128], fp32 via V_WMMA_F32_16X16X4_F32.
// One block = 8 waves = full 16x128 output tile (one 16x16 tile per wave).
// A tile staged through LDS with coalesced float4 loads.
// ------------------------------------------------------------------
__global__ __launch_bounds__(256) void gemm128_wmma(
    const float* __restrict__ A, const float* __restrict__ W,
    float* __restrict__ H, int N)
{
    __shared__ float As[16 * 128];   // 8 KB
    const int rowBase = blockIdx.x << 4;

    // Cooperative, coalesced load of the 16x128 A tile (512 float4s).
    for (int i = threadIdx.x; i < 512; i += 256) {
        const int rrow = i >> 5;          // 32 float4s per 128-float row
        const int rcol = i & 31;
        const int gr   = rowBase + rrow;
        float4 v = make_float4(0.f, 0.f, 0.f, 0.f);
        if (gr < N)
            v = reinterpret_cast<const float4*>(A + (size_t)gr * 128)[rcol];
        reinterpret_cast<float4*>(As)[i] = v;
    }
    __syncthreads();

    const int lane    = threadIdx.x & 31;
    const int wave    = threadIdx.x >> 5;   // 0..7 -> column tile
    const int colBase = wave << 4;
    const int r       = lane & 15;          // M row (A frag) / N col (B frag)
    const int kh      = lane >> 4;          // K-pair select (ISA 16x4 f32 layout)

    v8f acc = {};
    #pragma unroll
    for (int k = 0; k < 128; k += 4) {
        const int ka = k + 2 * kh;
        v2f a, b;
        // A fragment: VGPR0 = A[M=r][ka], VGPR1 = A[M=r][ka+1]
        a.x = As[r * 128 + ka];
        a.y = As[r * 128 + ka + 1];
        // B fragment: VGPR0 = W[ka][col], VGPR1 = W[ka+1][col] (W stays in L2)
        b.x = W[(size_t)ka * 128 + colBase + r];
        b.y = W[(size_t)(ka + 1) * 128 + colBase + r];
        acc = __builtin_amdgcn_wmma_f32_16x16x4_f32(false, a, false, b,
                                                    (short)0, acc, false, false);
    }

    // C/D layout: VGPR v -> M = v + 8*kh, N = r
    #pragma unroll
    for (int v = 0; v < 8; ++v) {
        const int gr = rowBase + 8 * kh + v;
        if (gr < N) H[(size_t)gr * 128 + colBase + r] = acc[v];
    }
}

// ------------------------------------------------------------------
// Zero a float buffer (float4 stores).
// ------------------------------------------------------------------
__global__ __launch_bounds__(256) void zero_f32(float* __restrict__ p, int n4)
{
    int i = blockIdx.x * 256 + threadIdx.x;
    if (i < n4)
        reinterpret_cast<float4*>(p)[i] = make_float4(0.f, 0.f, 0.f, 0.f);
}

// ------------------------------------------------------------------
// Edge scatter: AGG[dst[e]] += H[src[e]] * ew[e].
// One wave per edge; each lane handles 4 channels (float4 gather,
// 4x native global_atomic_add_f32).
// ------------------------------------------------------------------
__global__ __launch_bounds__(256) void scatter_edges(
    const float* __restrict__ H,
    const int*   __restrict__ src,
    const int*   __restrict__ dst,
    const float* __restrict__ ew,
    float* __restrict__ AGG, int E)
{
    const int e = blockIdx.x * 8 + (threadIdx.x >> 5);
    if (e >= E) return;
    const int lane = threadIdx.x & 31;

    const int   s = src[e];
    const int   d = dst[e];
    const float w = ew[e];

    float4 m = reinterpret_cast<const float4*>(H + (size_t)s * 128)[lane];
    float* out = AGG + (size_t)d * 128 + lane * 4;
    unsafeAtomicAdd(out + 0, m.x * w);
    unsafeAtomicAdd(out + 1, m.y * w);
    unsafeAtomicAdd(out + 2, m.z * w);
    unsafeAtomicAdd(out + 3, m.w * w);
}

// ------------------------------------------------------------------
// Fused bias + LayerNorm + ReLU over 128-wide rows.
// One wave per row; wave32 reductions with __shfl_xor. Safe in-place.
// ------------------------------------------------------------------
__global__ __launch_bounds__(256) void bias_ln_relu(
    const float* __restrict__ AGG,
    const float* __restrict__ bias,
    const float* __restrict__ gamma,
    const float* __restrict__ beta,
    float* __restrict__ out, int N)
{
    const int n = blockIdx.x * 8 + (threadIdx.x >> 5);
    if (n >= N) return;
    const int lane = threadIdx.x & 31;

    float4 v = reinterpret_cast<const float4*>(AGG + (size_t)n * 128)[lane];
    float4 b = reinterpret_cast<const float4*>(bias)[lane];
    v.x += b.x; v.y += b.y; v.z += b.z; v.w += b.w;

    float s = v.x + v.y + v.z + v.w;
    #pragma unroll
    for (int off = 16; off >= 1; off >>= 1) s += __shfl_xor(s, off, 32);
    const float mu = s * (1.0f / 128.0f);

    const float dx = v.x - mu, dy = v.y - mu, dz = v.z - mu, dw = v.w - mu;
    float q = dx * dx + dy * dy + dz * dz + dw * dw;
    #pragma unroll
    for (int off = 16; off >= 1; off >>= 1) q += __shfl_xor(q, off, 32);
    const float rstd = rsqrtf(q * (1.0f / 128.0f) + LN_EPS);

    float4 g  = reinterpret_cast<const float4*>(gamma)[lane];
    float4 be = reinterpret_cast<const float4*>(beta)[lane];
    float4 o;
    o.x = fmaxf(dx * rstd * g.x + be.x, 0.f);
    o.y = fmaxf(dy * rstd * g.y + be.y, 0.f);
    o.z = fmaxf(dz * rstd * g.z + be.z, 0.f);
    o.w = fmaxf(dw * rstd * g.w + be.w, 0.f);
    reinterpret_cast<float4*>(out + (size_t)n * 128)[lane] = o;
}

// ------------------------------------------------------------------
// Host-side launcher.
// Inputs (setup_inputs order):
//  0:x 1:edge_index 2:edge_weight 3:W1 4:b1 5:W2 6:b2 7:g1 8:be1 9:g2 10:be2
// Buffer plan (only 25.6 MB of ws needed):
//  gemm1: x -> ws ; zero(out) ; scatter: ws -> out ; LN1 in-place(out)
//  gemm2: out -> ws ; zero(out) ; scatter: ws -> out ; LN2 in-place(out)
// ------------------------------------------------------------------
extern "C" void kernel_launch(void* const* d_in, const int* in_sizes, int n_in,
                              void* d_out, int out_size, void* d_ws, size_t ws_size,
                              hipStream_t stream)
{
    const float* x   = (const float*)d_in[0];
    const int*   ei  = (const int*)  d_in[1];
    const float* ew  = (const float*)d_in[2];
    const float* W1  = (const float*)d_in[3];
    const float* b1  = (const float*)d_in[4];
    const float* W2  = (const float*)d_in[5];
    const float* b2  = (const float*)d_in[6];
    const float* g1  = (const float*)d_in[7];
    const float* be1 = (const float*)d_in[8];
    const float* g2  = (const float*)d_in[9];
    const float* be2 = (const float*)d_in[10];

    const int N = in_sizes[0] / 128;
    const int E = in_sizes[2];
    const int* src = ei;
    const int* dst = ei + E;

    float* H   = (float*)d_ws;    // N*128 floats (25.6 MB)
    float* out = (float*)d_out;   // doubles as aggregation buffer

    const int gemmBlocks = (N + 15) / 16;
    const int n4         = (N * 128) / 4;
    const int zeroBlocks = (n4 + 255) / 256;
    const int edgeBlocks = (E + 7) / 8;
    const int rowBlocks  = (N + 7) / 8;

    // ---- Layer 1 ----
    gemm128_wmma<<<gemmBlocks, 256, 0, stream>>>(x, W1, H, N);
    zero_f32<<<zeroBlocks, 256, 0, stream>>>(out, n4);
    scatter_edges<<<edgeBlocks, 256, 0, stream>>>(H, src, dst, ew, out, E);
    bias_ln_relu<<<rowBlocks, 256, 0, stream>>>(out, b1, g1, be1, out, N);

    // ---- Layer 2 ----
    gemm128_wmma<<<gemmBlocks, 256, 0, stream>>>(out, W2, H, N);
    zero_f32<<<zeroBlocks, 256, 0, stream>>>(out, n4);
    scatter_edges<<<edgeBlocks, 256, 0, stream>>>(H, src, dst, ew, out, E);
    bias_ln_relu<<<rowBlocks, 256, 0, stream>>>(out, b2, g2, be2, out, N);
}